// StageNetLayer_9912784519822
// MI455X (gfx1250) — compile-verified
//
#include <hip/hip_runtime.h>
#include <hip/hip_bf16.h>

// ---------------- problem constants (from reference) ----------------
constexpr int Bc  = 512;                // batch
constexpr int Tt  = 128;                // time steps
constexpr int Dd  = 256;                // input dim
constexpr int Ll  = 3;                  // levels
constexpr int CHc = 128;                // channels per level
constexpr int CSc = 10;                 // conv window / history ring
constexpr int Hh  = Ll * CHc;           // 384
constexpr int Gg  = 4 * Hh + 2 * Ll;    // 1542
constexpr int HSs = Hh / 6;             // 64

// ---------------- kernel tiling ----------------
constexpr int M       = 16;             // batch tile (WMMA M)
constexpr int NBLK    = Bc / M;         // 32 workgroups
constexpr int NTHREAD = 512;            // 16 wave32s
constexpr int NWAVES  = NTHREAD / 32;
constexpr int HPAD    = 388;            // H + ti col, padded to mult of 4
constexpr int KXP     = 260;            // D+1 padded to mult of 4
constexpr int XOPAD   = 1552;           // G padded to 97*16
constexpr int NT_G    = XOPAD / 16;     // 97 N-tiles for xo GEMM
constexpr int NT_H    = Hh / 16;        // 24 N-tiles for H-wide GEMMs
constexpr int REP     = (M * Hh) / NTHREAD; // 12 elements/thread in M*H passes

// ---------------- workspace layout (floats) ----------------
constexpr size_t WS_RING = 0;                                   // NBLK*CS*M*H
constexpr size_t WS_KWT  = WS_RING + (size_t)NBLK * CSc * M * Hh; // [KXP][XOPAD]
constexpr size_t WS_RWT  = WS_KWT + (size_t)KXP * XOPAD;          // [HPAD][XOPAD]
constexpr size_t WS_CWT  = WS_RWT + (size_t)HPAD * XOPAD;         // [CS][H][H]
constexpr size_t WS_SWT  = WS_CWT + (size_t)CSc * Hh * Hh;        // [H][HS]
constexpr size_t WS_RSWT = WS_SWT + (size_t)Hh * HSs;             // [HS][H]
constexpr size_t WS_TOTAL = WS_RSWT + (size_t)HSs * Hh;           // ~18 MB

// ---------------- LDS layout (floats) ----------------
constexpr int SM_H    = 0;                       // [M][HPAD] h | ti | 0-pad
constexpr int SM_C    = SM_H  + M * HPAD;        // [M][Hh]   cell state
constexpr int SM_XO   = SM_C  + M * Hh;          // [M][XOPAD]
constexpr int SM_FM   = SM_XO + M * XOPAD;       // [M][4]
constexpr int SM_IM   = SM_FM + M * 4;           // [M][4]
constexpr int SM_RD   = SM_IM + M * 4;           // [CS][M] dis ring
constexpr int SM_DIS  = SM_RD + CSc * M;         // [M][CS] local_dis
constexpr int SM_LH   = SM_DIS + M * CSc;        // [M][Hh] staged local_h slice
constexpr int SM_TIN  = SM_LH + M * Hh;          // [M][Hh] theme input
constexpr int SM_MID  = SM_TIN + M * Hh;         // [M][HS]
constexpr int SM_TH   = SM_MID + M * HSs;        // [M][Hh] theme gate
constexpr int SMEM_FLOATS = SM_TH + M * Hh;      // 57088 floats = 228,352 B

typedef float v2f __attribute__((ext_vector_type(2)));
typedef float v8f __attribute__((ext_vector_type(8)));

__device__ __forceinline__ v8f wmma_f32(v2f a, v2f b, v8f c) {
    // V_WMMA_F32_16X16X4_F32 : D = A(16x4) * B(4x16) + C(16x16), full fp32
    return __builtin_amdgcn_wmma_f32_16x16x4_f32(false, a, false, b,
                                                 (short)0, c, false, false);
}

__device__ __forceinline__ float sigf(float v) { return 1.0f / (1.0f + __expf(-v)); }

__global__ __launch_bounds__(NTHREAD)
void stagenet_scan_kernel(const float* __restrict__ x,     // (B,T,D)
                          const float* __restrict__ timep, // (B,T)
                          const float* __restrict__ kW,    // (G, D+1)
                          const float* __restrict__ kb,    // (G)
                          const float* __restrict__ rW,    // (G, H+1)
                          const float* __restrict__ rb,    // (G)
                          const float* __restrict__ sW,    // (HS, H)
                          const float* __restrict__ sb,    // (HS)
                          const float* __restrict__ rsW,   // (H, HS)
                          const float* __restrict__ rsb,   // (H)
                          const float* __restrict__ cW,    // (H, H, CS)
                          const float* __restrict__ cb,    // (H)
                          float* __restrict__ out_last,    // (B,H)
                          float* __restrict__ out_seq,     // (B,T,H)
                          float* __restrict__ out_dist,    // (T,B)
                          float* __restrict__ ws)          // workspace
{
    extern __shared__ float sm[];
    float* sh_h   = sm + SM_H;
    float* sh_c   = sm + SM_C;
    float* sh_xo  = sm + SM_XO;
    float* sh_fm  = sm + SM_FM;
    float* sh_im  = sm + SM_IM;
    float* sh_rd  = sm + SM_RD;
    float* sh_dis = sm + SM_DIS;
    float* sh_lh  = sm + SM_LH;
    float* sh_tin = sm + SM_TIN;
    float* sh_mid = sm + SM_MID;
    float* sh_th  = sm + SM_TH;

    const int tid   = threadIdx.x;
    const int lane  = tid & 31;
    const int w     = tid >> 5;
    const int lhalf = lane >> 4;      // K-pair half / M-half of fragment
    const int lmod  = lane & 15;      // fragment row (A) / col (B, C)
    const int bid   = blockIdx.x;
    const int b0    = bid * M;

    float* ring = ws + WS_RING;
    float* kWt  = ws + WS_KWT;        // [k][n] zero-padded
    float* rWt  = ws + WS_RWT;        // [k][n] zero-padded
    float* cWt  = ws + WS_CWT;        // [k][c][o]
    float* sWt  = ws + WS_SWT;        // [c][s]
    float* rsWt = ws + WS_RSWT;       // [s][j]
    float* blk_ring = ring + (size_t)bid * CSc * M * Hh;

    // ---- one-time weight transpose into padded, N-coalesced layouts ----
    // (every block writes identical values: benign; each block fences its own)
    for (int i = tid; i < KXP * XOPAD; i += NTHREAD) {
        const int k = i / XOPAD, n = i - k * XOPAD;
        kWt[i] = (n < Gg && k < Dd + 1) ? kW[(size_t)n * (Dd + 1) + k] : 0.0f;
    }
    for (int i = tid; i < HPAD * XOPAD; i += NTHREAD) {
        const int k = i / XOPAD, n = i - k * XOPAD;
        rWt[i] = (n < Gg && k < Hh + 1) ? rW[(size_t)n * (Hh + 1) + k] : 0.0f;
    }
    for (int i = tid; i < CSc * Hh * Hh; i += NTHREAD) {
        const int k = i / (Hh * Hh), rem = i - k * Hh * Hh;
        const int c = rem / Hh, o = rem - c * Hh;
        cWt[i] = cW[((size_t)o * Hh + c) * CSc + k];
    }
    for (int i = tid; i < Hh * HSs; i += NTHREAD) {
        const int c = i / HSs, s = i - c * HSs;
        sWt[i] = sW[(size_t)s * Hh + c];
    }
    for (int i = tid; i < HSs * Hh; i += NTHREAD) {
        const int s = i / Hh, j = i - s * Hh;
        rsWt[i] = rsW[(size_t)j * HSs + s];
    }
    // ---- zero-init recurrent state ----
    for (int i = tid; i < CSc * M * Hh; i += NTHREAD) blk_ring[i] = 0.0f;
    for (int i = tid; i < M * HPAD;     i += NTHREAD) sh_h[i] = 0.0f;
    for (int i = tid; i < M * Hh;       i += NTHREAD) sh_c[i] = 0.0f;
    for (int i = tid; i < CSc * M;      i += NTHREAD) sh_rd[i] = 0.0f;
    __threadfence();
    __syncthreads();

    for (int t = 0; t < Tt; ++t) {
        // ------- stage ti into the h operand (col Hh) -------
        if (tid < M)
            sh_h[tid * HPAD + Hh] = timep[(size_t)(b0 + tid) * Tt + t];
        __syncthreads();

        // ===== GEMM 1: xo = [x,ti]@kW^T + kb + [h,ti]@rW^T + rb =====
        {
            const float* xrow = x + ((size_t)(b0 + lmod) * Tt + t) * Dd;
            const float  tival = sh_h[lmod * HPAD + Hh];
            for (int nt = w; nt < NT_G; nt += NWAVES) {
                const int n = nt * 16 + lmod;
                v8f acc = {};
                // x part: K in [0,256), branch-free, v2f A loads, coalesced B
                #pragma unroll 4
                for (int k0 = 0; k0 < Dd; k0 += 4) {
                    const int kk = k0 + 2 * lhalf;
                    const v2f a = *(const v2f*)(xrow + kk);
                    v2f b;
                    b.x = kWt[(size_t)kk * XOPAD + n];
                    b.y = kWt[(size_t)(kk + 1) * XOPAD + n];
                    acc = wmma_f32(a, b, acc);
                }
                {   // tail: kk in {256..259}; col 256 = ti, rest zero (kWt padded)
                    const int kk = Dd + 2 * lhalf;
                    v2f a; a.x = (lhalf == 0) ? tival : 0.0f; a.y = 0.0f;
                    v2f b;
                    b.x = kWt[(size_t)kk * XOPAD + n];
                    b.y = kWt[(size_t)(kk + 1) * XOPAD + n];
                    acc = wmma_f32(a, b, acc);
                }
                // h part: K in [0,388); LDS pad cols and rWt pad rows are zero
                #pragma unroll 4
                for (int k0 = 0; k0 < HPAD; k0 += 4) {
                    const int kk = k0 + 2 * lhalf;
                    const v2f a = *(const v2f*)(sh_h + lmod * HPAD + kk);
                    v2f b;
                    b.x = rWt[(size_t)kk * XOPAD + n];
                    b.y = rWt[(size_t)(kk + 1) * XOPAD + n];
                    acc = wmma_f32(a, b, acc);
                }
                const float bias = (n < Gg) ? (kb[n] + rb[n]) : 0.0f;
                #pragma unroll
                for (int r = 0; r < 8; ++r)
                    sh_xo[(r + 8 * lhalf) * XOPAD + n] = acc[r] + bias;
            }
        }
        __syncthreads();

        // ===== cumax gates (L=3), dis ring =====
        if (tid < M) {
            const int b = tid;
            const float* xb = sh_xo + b * XOPAD;
            float z0 = xb[0], z1 = xb[1], z2 = xb[2];
            float mx = fmaxf(z0, fmaxf(z1, z2));
            float e0 = __expf(z0 - mx), e1 = __expf(z1 - mx), e2 = __expf(z2 - mx);
            float inv = 1.0f / (e0 + e1 + e2);
            float f0 = e0 * inv, f1 = f0 + e1 * inv, f2 = f1 + e2 * inv;
            sh_fm[b * 4 + 0] = f0; sh_fm[b * 4 + 1] = f1; sh_fm[b * 4 + 2] = f2;
            z0 = xb[3]; z1 = xb[4]; z2 = xb[5];
            mx = fmaxf(z0, fmaxf(z1, z2));
            e0 = __expf(z0 - mx); e1 = __expf(z1 - mx); e2 = __expf(z2 - mx);
            inv = 1.0f / (e0 + e1 + e2);
            float p0 = e0 * inv, p1 = e1 * inv, p2 = e2 * inv;
            sh_im[b * 4 + 2] = p2; sh_im[b * 4 + 1] = p2 + p1; sh_im[b * 4 + 0] = p2 + p1 + p0;
            const float cur = 1.0f - (f0 + f1 + f2) * (1.0f / 3.0f);
            sh_rd[(t % CSc) * M + b] = cur;
            out_dist[(size_t)t * Bc + b0 + b] = cur;
        }
        __syncthreads();

        // ===== pointwise cell update =====
        #pragma unroll
        for (int r = 0; r < REP; ++r) {
            const int idx = tid + r * NTHREAD;
            const int b = idx / Hh, j = idx - b * Hh;
            const int l = j >> 7, ch = j & (CHc - 1);
            const float* gb = sh_xo + b * XOPAD + 2 * Ll;
            const float fg = sigf(gb[(0 + l) * CHc + ch]);
            const float ig = sigf(gb[(3 + l) * CHc + ch]);
            const float og = sigf(gb[(6 + l) * CHc + ch]);
            const float ci = tanhf(gb[(9 + l) * CHc + ch]);
            const float fm = sh_fm[b * 4 + l], im = sh_im[b * 4 + l];
            const float cl = sh_c[b * Hh + j];
            const float ov = fm * im;
            const float cn = ov * (fg * cl + ig * ci) + (fm - ov) * cl + (im - ov) * ci;
            const float hn = og * tanhf(cn);
            sh_c[b * Hh + j] = cn;
            sh_h[b * HPAD + j] = hn;
            blk_ring[(t % CSc) * M * Hh + idx] = hn;   // history ring (L2-resident)
        }
        __threadfence_block();
        __syncthreads();

        // ===== local_dis = softmax(cumsum(dis history)) =====
        const int tb = (t + 1) % CSc;                 // chronological base slot
        if (tid < M) {
            const int b = tid;
            float v[CSc], cum = 0.0f;
            #pragma unroll
            for (int k = 0; k < CSc; ++k) {
                int s = tb + k; if (s >= CSc) s -= CSc;
                cum += sh_rd[s * M + b];
                v[k] = cum;
            }
            float mx = v[0];
            #pragma unroll
            for (int k = 1; k < CSc; ++k) mx = fmaxf(mx, v[k]);
            float sum = 0.0f;
            #pragma unroll
            for (int k = 0; k < CSc; ++k) { v[k] = __expf(v[k] - mx); sum += v[k]; }
            const float inv = 1.0f / sum;
            #pragma unroll
            for (int k = 0; k < CSc; ++k) sh_dis[b * CSc + k] = v[k] * inv;
        }
        __syncthreads();

        // ===== conv GEMM, k-major: stage local_h slice in LDS per k =====
        // also accumulates theme_in (mean over k) in registers for free
        const int nA = w * 16 + lmod;                 // tiles 0..15
        const bool hasB = (w + NWAVES) < NT_H;        // tiles 16..23 (waves 0..7)
        const int nB = (w + NWAVES) * 16 + lmod;
        v8f accA = {}, accB = {};
        float tin_acc[REP];
        #pragma unroll
        for (int r = 0; r < REP; ++r) tin_acc[r] = 0.0f;

        for (int k = 0; k < CSc; ++k) {
            int sl = tb + k; if (sl >= CSc) sl -= CSc;
            const float* rk = blk_ring + (size_t)sl * M * Hh;
            #pragma unroll
            for (int r = 0; r < REP; ++r) {
                const int idx = tid + r * NTHREAD;
                const int b = idx / Hh;
                const float v = rk[idx] * sh_dis[b * CSc + k];
                sh_lh[idx] = v;
                tin_acc[r] += v;
            }
            __syncthreads();
            const float* bkc = cWt + (size_t)k * Hh * Hh;   // [c][o]
            const float* arow = sh_lh + lmod * Hh;
            #pragma unroll 4
            for (int c0 = 0; c0 < Hh; c0 += 4) {
                const int c = c0 + 2 * lhalf;
                const v2f a = *(const v2f*)(arow + c);
                v2f bA;
                bA.x = bkc[(size_t)c * Hh + nA];
                bA.y = bkc[(size_t)(c + 1) * Hh + nA];
                accA = wmma_f32(a, bA, accA);
                if (hasB) {
                    v2f bB;
                    bB.x = bkc[(size_t)c * Hh + nB];
                    bB.y = bkc[(size_t)(c + 1) * Hh + nB];
                    accB = wmma_f32(a, bB, accB);
                }
            }
            __syncthreads();
        }
        // theme input = mean over k of local_h
        #pragma unroll
        for (int r = 0; r < REP; ++r)
            sh_tin[tid + r * NTHREAD] = tin_acc[r] * (1.0f / CSc);
        __syncthreads();

        // ===== theme GEMM A: (M,H)x(H,HS) + relu =====
        if (w < HSs / 16) {  // 4 waves, one 16-col tile each
            const int n = w * 16 + lmod;
            v8f acc = {};
            #pragma unroll 4
            for (int k0 = 0; k0 < Hh; k0 += 4) {
                const int kk = k0 + 2 * lhalf;
                const v2f a = *(const v2f*)(sh_tin + lmod * Hh + kk);
                v2f b;
                b.x = sWt[(size_t)kk * HSs + n];
                b.y = sWt[(size_t)(kk + 1) * HSs + n];
                acc = wmma_f32(a, b, acc);
            }
            const float bias = sb[n];
            #pragma unroll
            for (int r = 0; r < 8; ++r)
                sh_mid[(r + 8 * lhalf) * HSs + n] = fmaxf(acc[r] + bias, 0.0f);
        }
        __syncthreads();

        // ===== theme GEMM B: (M,HS)x(HS,H) + sigmoid =====
        for (int nt = w; nt < NT_H; nt += NWAVES) {
            const int n = nt * 16 + lmod;
            v8f acc = {};
            #pragma unroll 4
            for (int k0 = 0; k0 < HSs; k0 += 4) {
                const int kk = k0 + 2 * lhalf;
                const v2f a = *(const v2f*)(sh_mid + lmod * HSs + kk);
                v2f b;
                b.x = rsWt[(size_t)kk * Hh + n];
                b.y = rsWt[(size_t)(kk + 1) * Hh + n];
                acc = wmma_f32(a, b, acc);
            }
            const float bias = rsb[n];
            #pragma unroll
            for (int r = 0; r < 8; ++r)
                sh_th[(r + 8 * lhalf) * Hh + n] = sigf(acc[r] + bias);
        }
        __syncthreads();

        // ===== epilogue: out = theme * (conv + cb) + h =====
        {
            const float cA = cb[nA];
            #pragma unroll
            for (int r = 0; r < 8; ++r) {
                const int m = r + 8 * lhalf;
                const float val = sh_th[m * Hh + nA] * (accA[r] + cA) + sh_h[m * HPAD + nA];
                out_seq[((size_t)(b0 + m) * Tt + t) * Hh + nA] = val;
                if (t == Tt - 1) out_last[(size_t)(b0 + m) * Hh + nA] = val;
            }
            if (hasB) {
                const float cB = cb[nB];
                #pragma unroll
                for (int r = 0; r < 8; ++r) {
                    const int m = r + 8 * lhalf;
                    const float val = sh_th[m * Hh + nB] * (accB[r] + cB) + sh_h[m * HPAD + nB];
                    out_seq[((size_t)(b0 + m) * Tt + t) * Hh + nB] = val;
                    if (t == Tt - 1) out_last[(size_t)(b0 + m) * Hh + nB] = val;
                }
            }
        }
        __syncthreads();
    }
}

extern "C" void kernel_launch(void* const* d_in, const int* in_sizes, int n_in,
                              void* d_out, int out_size, void* d_ws, size_t ws_size,
                              hipStream_t stream) {
    (void)in_sizes; (void)n_in; (void)out_size; (void)ws_size;
    const float* x    = (const float*)d_in[0];
    const float* tim  = (const float*)d_in[1];
    const float* kW   = (const float*)d_in[2];
    const float* kb   = (const float*)d_in[3];
    const float* rW   = (const float*)d_in[4];
    const float* rb   = (const float*)d_in[5];
    const float* sW   = (const float*)d_in[6];
    const float* sb   = (const float*)d_in[7];
    const float* rsW  = (const float*)d_in[8];
    const float* rsb  = (const float*)d_in[9];
    const float* cW   = (const float*)d_in[10];
    const float* cb   = (const float*)d_in[11];

    float* out      = (float*)d_out;
    float* out_last = out;                                  // (B,H)
    float* out_seq  = out_last + (size_t)Bc * Hh;           // (B,T,H)
    float* out_dist = out_seq + (size_t)Bc * Tt * Hh;       // (T,B)
    float* ws       = (float*)d_ws;                         // ~18 MB

    const size_t smem = (size_t)SMEM_FLOATS * sizeof(float);  // ~228 KB of 320 KB WGP LDS
    hipFuncSetAttribute((const void*)stagenet_scan_kernel,
                        hipFuncAttributeMaxDynamicSharedMemorySize, (int)smem);

    stagenet_scan_kernel<<<dim3(NBLK), dim3(NTHREAD), smem, stream>>>(
        x, tim, kW, kb, rW, rb, sW, sb, rsW, rsb, cW, cb,
        out_last, out_seq, out_dist, ws);
}